// LinearRoutingAttention_88519275970644
// MI455X (gfx1250) — compile-verified
//
#include <hip/hip_runtime.h>
#include <hip/hip_bf16.h>
#include <cstddef>

#define BTOK 1024
#define DMODEL 1024
#define DHEAD 64
#define NHEAD 16
#define NACT 4
#define EPSV 1e-6f

typedef __attribute__((ext_vector_type(2))) float v2f;
typedef __attribute__((ext_vector_type(4))) float v4f;
typedef __attribute__((ext_vector_type(8))) float v8f;

// ---------------------------------------------------------------------------
// Kernel 1: routing — logits = query @ We^T + be, softmax, top-4
// one block per token, 256 threads = 16 heads x 16 lanes
// ---------------------------------------------------------------------------
__global__ __launch_bounds__(256) void lra_routing(
    const float* __restrict__ query, const float* __restrict__ We,
    const float* __restrict__ be,
    int* __restrict__ top_i, float* __restrict__ top_p,
    float* __restrict__ top_logits)
{
    __shared__ float sq[DMODEL];
    __shared__ float slog[NHEAD];
    const int b = blockIdx.x;
    const int t = threadIdx.x;

    for (int i = t; i < DMODEL; i += 256) sq[i] = query[(size_t)b * DMODEL + i];
    __syncthreads();

    const int h = t >> 4;     // 0..15
    const int j = t & 15;     // 0..15
    const float* w = We + (size_t)h * DMODEL;
    float p = 0.f;
    for (int k = j; k < DMODEL; k += 16) p += sq[k] * w[k];
    // 16-lane reduction (groups of 16 lie inside one wave32)
    for (int off = 8; off > 0; off >>= 1) p += __shfl_xor(p, off, 16);
    if (j == 0) slog[h] = p + be[h];
    __syncthreads();

    if (t == 0) {
        float lmax = -1e30f;
        for (int i = 0; i < NHEAD; ++i) lmax = fmaxf(lmax, slog[i]);
        float ex[NHEAD]; float sum = 0.f;
        for (int i = 0; i < NHEAD; ++i) { ex[i] = expf(slog[i] - lmax); sum += ex[i]; }
        unsigned used = 0u;
        for (int a = 0; a < NACT; ++a) {
            int best = 0; float bv = -1e30f;
            for (int i = 0; i < NHEAD; ++i)
                if (!((used >> i) & 1u) && slog[i] > bv) { bv = slog[i]; best = i; }
            used |= 1u << best;
            top_i[b * NACT + a]      = best;
            top_p[b * NACT + a]      = ex[best] / sum;
            top_logits[b * NACT + a] = slog[best];
        }
    }
}

// ---------------------------------------------------------------------------
// Kernel 2: dense projection GEMM via V_WMMA_F32_16X16X4_F32
// C[M=1024 tokens, N=1024 feats] = X[M,K] * W[N,K]^T + bias[N]
// one wave per 16(M) x 64(N) tile; 4 accumulators share the A fragment.
// ---------------------------------------------------------------------------
__device__ __forceinline__ void lra_store_tile(float* __restrict__ C, v8f c,
                                               int mbase, int n, float bv)
{
#pragma unroll
    for (int r = 0; r < 8; ++r)
        C[(size_t)(mbase + r) * DMODEL + n] = c[r] + bv;
}

__global__ __launch_bounds__(32) void lra_proj_wmma(
    const float* __restrict__ X, const float* __restrict__ W,
    const float* __restrict__ bias, float* __restrict__ C)
{
    const int K = DMODEL;
    const int lane = threadIdx.x;
    const int half = lane >> 4;
    const int l16  = lane & 15;
    const int tileM = blockIdx.y;   // 0..63
    const int nblk  = blockIdx.x;   // 0..15 -> 64 output cols

    const float* xrow = X + ((size_t)(tileM * 16 + l16)) * K + 2 * half;
    const float* w0   = W + ((size_t)(nblk * 64 +  0 + l16)) * K + 2 * half;
    const float* w1   = w0 + (size_t)16 * K;
    const float* w2   = w0 + (size_t)32 * K;
    const float* w3   = w0 + (size_t)48 * K;

    v8f c0 = {}; v8f c1 = {}; v8f c2 = {}; v8f c3 = {};
    for (int k = 0; k < K; k += 4) {
        v2f aa = *(const v2f*)(xrow + k);
        v2f b0 = *(const v2f*)(w0 + k);
        v2f b1 = *(const v2f*)(w1 + k);
        v2f b2 = *(const v2f*)(w2 + k);
        v2f b3 = *(const v2f*)(w3 + k);
        c0 = __builtin_amdgcn_wmma_f32_16x16x4_f32(false, aa, false, b0, (short)0, c0, false, false);
        c1 = __builtin_amdgcn_wmma_f32_16x16x4_f32(false, aa, false, b1, (short)0, c1, false, false);
        c2 = __builtin_amdgcn_wmma_f32_16x16x4_f32(false, aa, false, b2, (short)0, c2, false, false);
        c3 = __builtin_amdgcn_wmma_f32_16x16x4_f32(false, aa, false, b3, (short)0, c3, false, false);
    }

    // C fragment: lanes 0-15 -> rows M=r, lanes 16-31 -> rows M=8+r, col = l16
    const int mbase = tileM * 16 + half * 8;
    int n0 = nblk * 64 + l16;
    lra_store_tile(C, c0, mbase, n0,      bias[n0]);
    lra_store_tile(C, c1, mbase, n0 + 16, bias[n0 + 16]);
    lra_store_tile(C, c2, mbase, n0 + 32, bias[n0 + 32]);
    lra_store_tile(C, c3, mbase, n0 + 48, bias[n0 + 48]);
}

// ---------------------------------------------------------------------------
// Kernel 3: per-(b,h) state pass, b128 + non-temporal on the 536MB S stream.
// Unselected heads: NT float4 copy. Selected heads: linear-attention read +
// gated write with the S tile held in registers (read once, write once).
// grid = B*H blocks of 256 threads.
// float4 index i4 = t + 256j  ->  row d = (t>>4) + 16j, cols e = 4*(t&15)..+3
// ---------------------------------------------------------------------------
__global__ __launch_bounds__(256) void lra_state(
    const float* __restrict__ matrix, const float* __restrict__ normalizer,
    const float* __restrict__ qf, const float* __restrict__ kf,
    const float* __restrict__ vf,
    const int* __restrict__ top_i, const float* __restrict__ top_p,
    const float* __restrict__ top_logits,
    const float* __restrict__ Ww, const float* __restrict__ bw,
    const float* __restrict__ decay_logits,
    float* __restrict__ matrix_new, float* __restrict__ normalizer_new,
    float* __restrict__ outp)
{
    const int bh = blockIdx.x;
    const int b = bh >> 4;
    const int h = bh & 15;
    const int t = threadIdx.x;

    __shared__ int sa;
    if (t == 0) {
        int a = -1;
        for (int i = 0; i < NACT; ++i) if (top_i[b * NACT + i] == h) a = i;
        sa = a;
    }
    __syncthreads();
    const int a = sa;

    const v4f* S4  = (const v4f*)(matrix     + (size_t)bh * (DHEAD * DHEAD));
    v4f*       So4 = (v4f*)(matrix_new       + (size_t)bh * (DHEAD * DHEAD));
    const float* Z  = normalizer     + (size_t)bh * DHEAD;
    float*       Zo = normalizer_new + (size_t)bh * DHEAD;

    if (a < 0) {                      // head not selected: NT streaming copy
#pragma unroll
        for (int j = 0; j < 4; ++j) {
            v4f val = __builtin_nontemporal_load(S4 + t + 256 * j);
            __builtin_nontemporal_store(val, So4 + t + 256 * j);
        }
        if (t < DHEAD) Zo[t] = Z[t];
        return;
    }

    __shared__ __align__(16) float s_q[DHEAD], s_phiq[DHEAD], s_phik[DHEAD], s_v[DHEAD];
    __shared__ __align__(16) float s_num[DHEAD], s_out[DHEAD], s_fac[DHEAD], s_wk[DHEAD];
    __shared__ v4f   s_red4[256];
    __shared__ float s_red[128];
    __shared__ float s_scal[2];
    __shared__ float s_wp;

    if (t < DHEAD) {
        float q = qf[(size_t)b * DMODEL + h * DHEAD + t];
        float k = kf[(size_t)b * DMODEL + h * DHEAD + t];
        float v = vf[(size_t)b * DMODEL + h * DHEAD + t];
        s_q[t]    = q;
        s_phiq[t] = (q > 0.f) ? q + 1.f : expf(q);   // elu(x)+1
        s_phik[t] = (k > 0.f) ? k + 1.f : expf(k);
        s_v[t]    = v;
    }
    __syncthreads();

    const int rrow = t >> 4;          // 0..15 (row-phase within column group)
    const int e0   = (t & 15) * 4;    // first of 4 owned columns
    v4f sreg[4];
    v4f pn = {};
#pragma unroll
    for (int j = 0; j < 4; ++j) {
        v4f sv = __builtin_nontemporal_load(S4 + t + 256 * j);
        sreg[j] = sv;
        float pq = s_phiq[rrow + 16 * j];
        pn.x += pq * sv.x; pn.y += pq * sv.y; pn.z += pq * sv.z; pn.w += pq * sv.w;
    }
    s_red4[t] = pn;
    __syncthreads();
    if (t < DHEAD) {                  // num[e] = sum over the 16 row-phases
        int g = t >> 2, c = t & 3;
        float acc = 0.f;
        for (int r = 0; r < 16; ++r)
            acc += ((const float*)&s_red4[r * 16 + g])[c];
        s_num[t] = acc;
    }
    __syncthreads();

    if (t < DHEAD) {
        s_red[t]      = s_phiq[t] * s_phik[t];
        s_red[t + 64] = s_phiq[t] * (Z[t] + s_phik[t]);
    }
    __syncthreads();
    if (t == 0) {
        float qk = 0.f, den = 0.f;
        for (int i = 0; i < DHEAD; ++i) { qk += s_red[i]; den += s_red[64 + i]; }
        s_scal[0] = qk;
        s_scal[1] = den + EPSV;
    }
    __syncthreads();
    const float qk  = s_scal[0];
    const float den = s_scal[1];

    if (t < DHEAD)
        s_out[t] = (s_num[t] + qk * s_v[t]) / den;
    __syncthreads();

    if (t < DHEAD) s_red[t] = (s_out[t] + s_q[t]) * Ww[h * DHEAD + t];
    __syncthreads();
    if (t == 0) {
        float acc = 0.f;
        for (int i = 0; i < DHEAD; ++i) acc += s_red[i];
        float wl = top_logits[b * NACT + a] + acc + bw[h];
        s_wp = 1.f / (1.f + expf(-wl));
    }
    __syncthreads();
    const float wp = s_wp;

    if (t < DHEAD) {
        float dec = 1.f / (1.f + expf(-decay_logits[h * DHEAD + t]));
        s_fac[t] = 1.f - wp * dec;
        s_wk[t]  = wp * s_phik[t];
        Zo[t] = Z[t] * s_fac[t] + s_wk[t];
        outp[((size_t)b * NACT + a) * DHEAD + t] = s_out[t] * top_p[b * NACT + a];
    }
    __syncthreads();

    const v4f vv = *(const v4f*)&s_v[e0];
#pragma unroll
    for (int j = 0; j < 4; ++j) {
        int d = rrow + 16 * j;
        float f = s_fac[d], wk = s_wk[d];
        v4f sv = sreg[j];
        v4f o;
        o.x = sv.x * f + wk * vv.x;
        o.y = sv.y * f + wk * vv.y;
        o.z = sv.z * f + wk * vv.z;
        o.w = sv.w * f + wk * vv.w;
        __builtin_nontemporal_store(o, So4 + t + 256 * j);
    }
}

// ---------------------------------------------------------------------------
// Kernel 4: merged[b,m] = sum_a ( Wo[h_a][m,:] . outp[b,a,:] + p_a * bo[h_a,m] )
// grid = (DMODEL/256, B), block 256; Wo rows read as b128 (L2-resident reuse)
// ---------------------------------------------------------------------------
__global__ __launch_bounds__(256) void lra_merge(
    const float* __restrict__ outp, const int* __restrict__ top_i,
    const float* __restrict__ top_p, const float* __restrict__ Wo,
    const float* __restrict__ bo, float* __restrict__ merged)
{
    const int b = blockIdx.y;
    const int m = blockIdx.x * 256 + threadIdx.x;
    const int t = threadIdx.x;

    __shared__ __align__(16) float so[NACT * DHEAD];
    __shared__ int   sh[NACT];
    __shared__ float sp[NACT];
    so[t] = outp[(size_t)b * (NACT * DHEAD) + t];
    if (t < NACT) { sh[t] = top_i[b * NACT + t]; sp[t] = top_p[b * NACT + t]; }
    __syncthreads();

    float acc = 0.f;
#pragma unroll
    for (int a = 0; a < NACT; ++a) {
        const int hh = sh[a];
        const v4f* w4 = (const v4f*)(Wo + ((size_t)hh * DMODEL + m) * DHEAD);
        const v4f* o4 = (const v4f*)&so[a * DHEAD];
        float s = 0.f;
#pragma unroll
        for (int ee = 0; ee < DHEAD / 4; ++ee) {
            v4f w = w4[ee], o = o4[ee];
            s += w.x * o.x + w.y * o.y + w.z * o.z + w.w * o.w;
        }
        acc += s + sp[a] * bo[(size_t)hh * DMODEL + m];
    }
    merged[(size_t)b * DMODEL + m] = acc;
}

// ---------------------------------------------------------------------------
// Host launcher
// ---------------------------------------------------------------------------
extern "C" void kernel_launch(void* const* d_in, const int* in_sizes, int n_in,
                              void* d_out, int out_size, void* d_ws, size_t ws_size,
                              hipStream_t stream)
{
    const float* query        = (const float*)d_in[0];
    const float* key          = (const float*)d_in[1];
    const float* value        = (const float*)d_in[2];
    const float* matrix       = (const float*)d_in[3];
    const float* normalizer   = (const float*)d_in[4];
    const float* Wq           = (const float*)d_in[5];
    const float* bq           = (const float*)d_in[6];
    const float* Wk           = (const float*)d_in[7];
    const float* bk           = (const float*)d_in[8];
    const float* Wv           = (const float*)d_in[9];
    const float* bv           = (const float*)d_in[10];
    const float* Wo           = (const float*)d_in[11];
    const float* bo           = (const float*)d_in[12];
    const float* We           = (const float*)d_in[13];
    const float* be           = (const float*)d_in[14];
    const float* Ww           = (const float*)d_in[15];
    const float* bw           = (const float*)d_in[16];
    const float* decay_logits = (const float*)d_in[17];

    // outputs: merged (B*DMODEL) | matrix_new (B*H*64*64) | normalizer_new (B*H*64)
    float* merged         = (float*)d_out;
    float* matrix_new     = merged + (size_t)BTOK * DMODEL;
    float* normalizer_new = matrix_new + (size_t)BTOK * NHEAD * DHEAD * DHEAD;

    // workspace carve-up
    char* ws = (char*)d_ws;
    int*   top_i      = (int*)ws;    ws += (size_t)BTOK * NACT * sizeof(int);
    float* top_p      = (float*)ws;  ws += (size_t)BTOK * NACT * sizeof(float);
    float* top_logits = (float*)ws;  ws += (size_t)BTOK * NACT * sizeof(float);
    float* qf         = (float*)ws;  ws += (size_t)BTOK * DMODEL * sizeof(float);
    float* kf         = (float*)ws;  ws += (size_t)BTOK * DMODEL * sizeof(float);
    float* vf         = (float*)ws;  ws += (size_t)BTOK * DMODEL * sizeof(float);
    float* outp       = (float*)ws;  ws += (size_t)BTOK * NACT * DHEAD * sizeof(float);

    // 1) routing
    lra_routing<<<BTOK, 256, 0, stream>>>(query, We, be, top_i, top_p, top_logits);

    // 2) dense projections (all heads) via f32 WMMA GEMM
    dim3 pg(DMODEL / 64, BTOK / 16);
    lra_proj_wmma<<<pg, 32, 0, stream>>>(query, Wq, bq, qf);
    lra_proj_wmma<<<pg, 32, 0, stream>>>(key,   Wk, bk, kf);
    lra_proj_wmma<<<pg, 32, 0, stream>>>(value, Wv, bv, vf);

    // 3) single-pass state copy/update (NT b128 streaming)
    lra_state<<<BTOK * NHEAD, 256, 0, stream>>>(
        matrix, normalizer, qf, kf, vf, top_i, top_p, top_logits,
        Ww, bw, decay_logits, matrix_new, normalizer_new, outp);

    // 4) merge through Wo
    dim3 mg(DMODEL / 256, BTOK);
    lra_merge<<<mg, 256, 0, stream>>>(outp, top_i, top_p, Wo, bo, merged);
}